// DGLCrossAttention_24678882083158
// MI455X (gfx1250) — compile-verified
//
#include <hip/hip_runtime.h>
#include <hip/hip_bf16.h>
#include <math.h>

#define HID 256
#define NHEAD 8
#define DK 32
// 1/sqrt(32)
#define SCALE_INV 0.17677669529663687f

typedef float v2f __attribute__((ext_vector_type(2)));
typedef float v8f __attribute__((ext_vector_type(8)));

// ---------------------------------------------------------------------------
// Core 16x16-tile fp32 WMMA GEMM step: acc += A(16x4) * B(4x16), K = 256.
// Lane layout (V_WMMA_F32_16X16X4_F32):
//   A: lane m = lane&15 holds row m, K = k0 + 2*(lane>>4) + {0,1}
//   B: lane n = lane&15 holds col n, same K pairing
//   D: lane&15 = col, VGPR r -> row r + 8*(lane>>4)
// ---------------------------------------------------------------------------
__device__ __forceinline__ v8f wmma_k256(const float* __restrict__ arow,
                                         const float* __restrict__ wrow,
                                         int khalf) {
    v8f acc = {};
    #pragma unroll 8
    for (int k0 = 0; k0 < HID; k0 += 4) {
        const float2 af = *(const float2*)(arow + k0 + khalf);
        const float2 bf = *(const float2*)(wrow + k0 + khalf);
        v2f a; a.x = af.x; a.y = af.y;
        v2f b; b.x = bf.x; b.y = bf.y;
        acc = __builtin_amdgcn_wmma_f32_16x16x4_f32(
            false, a, false, b, (short)0, acc, false, false);
    }
    return acc;
}

__device__ __forceinline__ void store_tile(float* __restrict__ Out, int row0,
                                           int n, int lane, const v8f& acc,
                                           float bval, int nrows) {
    if (row0 + 16 <= nrows) {           // uniform fast path: no exec juggling
        #pragma unroll
        for (int r = 0; r < 8; ++r) {
            const int M = r + ((lane >> 4) << 3);
            Out[(size_t)(row0 + M) * HID + n] = acc[r] + bval;
        }
    } else {
        #pragma unroll
        for (int r = 0; r < 8; ++r) {
            const int M = r + ((lane >> 4) << 3);
            if (row0 + M < nrows)
                Out[(size_t)(row0 + M) * HID + n] = acc[r] + bval;
        }
    }
}

// ---------------------------------------------------------------------------
// Fused Q/K/V projection: stage 16 input rows in LDS once, then three
// compile-time-unrolled GEMMs (uniform SGPR weight pointers -> global_load,
// no per-lane pointer selects). Each wave does 2 j-tiles per matrix.
// q gets bias bq; k, v are bias-free (matches reference).
// ---------------------------------------------------------------------------
__global__ __launch_bounds__(256) void qkv_proj_wmma(
    const float* __restrict__ In,
    const float* __restrict__ Wq, const float* __restrict__ Wk,
    const float* __restrict__ Wv, const float* __restrict__ bq,
    float* __restrict__ q, float* __restrict__ k, float* __restrict__ v,
    int nrows)
{
    __shared__ float lds_in[16 * HID];   // 16 KB of the 320 KB WGP LDS
    const int row0 = blockIdx.x * 16;

    for (int idx = threadIdx.x; idx < 16 * HID; idx += blockDim.x) {
        const int r = idx >> 8;
        const int c = idx & (HID - 1);
        const int gr = row0 + r;
        lds_in[idx] = (gr < nrows) ? In[(size_t)gr * HID + c] : 0.0f;
    }
    __syncthreads();

    const int wave  = threadIdx.x >> 5;
    const int lane  = threadIdx.x & 31;
    const int m     = lane & 15;
    const int khalf = (lane >> 4) << 1;
    const float* __restrict__ arow = &lds_in[m * HID];

    #pragma unroll
    for (int mat = 0; mat < 3; ++mat) {      // compile-time uniform matrix id
        const float* __restrict__ W = (mat == 0) ? Wq : (mat == 1) ? Wk : Wv;
        float* __restrict__       O = (mat == 0) ? q  : (mat == 1) ? k  : v;
        for (int jt = wave; jt < 16; jt += 8) {
            const int n = jt * 16 + m;
            const v8f acc = wmma_k256(arow, W + (size_t)n * HID, khalf);
            const float bval = (mat == 0) ? bq[n] : 0.0f;
            store_tile(O, row0, n, lane, acc, bval, nrows);
        }
    }
}

// ---------------------------------------------------------------------------
// Output projection with fused 1/z normalization of the LDS-staged rows:
// Out = (wv / z) @ Wo^T + bo
// ---------------------------------------------------------------------------
__global__ __launch_bounds__(256) void out_proj_wmma(
    const float* __restrict__ WV, const float* __restrict__ Z,
    const float* __restrict__ Wo, const float* __restrict__ bo,
    float* __restrict__ Out, int nrows)
{
    __shared__ float lds_in[16 * HID];
    const int row0 = blockIdx.x * 16;

    for (int idx = threadIdx.x; idx < 16 * HID; idx += blockDim.x) {
        const int r = idx >> 8;
        const int c = idx & (HID - 1);
        const int gr = row0 + r;
        float val = 0.0f;
        if (gr < nrows)
            val = WV[(size_t)gr * HID + c] / Z[(size_t)gr * NHEAD + (c >> 5)];
        lds_in[idx] = val;
    }
    __syncthreads();

    const int wave  = threadIdx.x >> 5;
    const int lane  = threadIdx.x & 31;
    const int m     = lane & 15;
    const int khalf = (lane >> 4) << 1;
    const float* __restrict__ arow = &lds_in[m * HID];

    for (int jt = wave; jt < 16; jt += 8) {
        const int n = jt * 16 + m;
        const v8f acc = wmma_k256(arow, Wo + (size_t)n * HID, khalf);
        store_tile(Out, row0, n, lane, acc, bo[n], nrows);
    }
}

// ---------------------------------------------------------------------------
// Edge attention: one wave32 per (edge, head).
// lane d: k[src,h,d]*q[dst,h,d] -> butterfly dot, exp(clip(./sqrt(dk))),
// scatter v[src,h,d]*score into wv[dst,h,d] via global_atomic_add_f32.
// ---------------------------------------------------------------------------
__global__ __launch_bounds__(256) void edge_attn(
    const float* __restrict__ q, const float* __restrict__ k,
    const float* __restrict__ v, const int* __restrict__ src,
    const int* __restrict__ dst, float* __restrict__ wv,
    float* __restrict__ z, int nedges)
{
    const int gid  = blockIdx.x * blockDim.x + threadIdx.x;
    const int pair = gid >> 5;
    const int lane = gid & 31;
    if (pair >= nedges * NHEAD) return;      // wave-uniform guard

    const int e = pair >> 3;
    const int h = pair & 7;
    const int s = src[e];
    const int d = dst[e];

    const size_t koff = (size_t)s * HID + h * DK + lane;
    const size_t qoff = (size_t)d * HID + h * DK + lane;

    float prod = k[koff] * q[qoff];
    #pragma unroll
    for (int off = 16; off > 0; off >>= 1)   // wave32 butterfly all-reduce
        prod += __shfl_xor(prod, off, 32);

    const float score = expf(fminf(fmaxf(prod * SCALE_INV, -10.0f), 10.0f));
    const float vv = v[koff];

    atomicAdd(&wv[qoff], vv * score);
    if (lane == 0)
        atomicAdd(&z[(size_t)d * NHEAD + h], score);
}

// ---------------------------------------------------------------------------
extern "C" void kernel_launch(void* const* d_in, const int* in_sizes, int n_in,
                              void* d_out, int out_size, void* d_ws, size_t ws_size,
                              hipStream_t stream) {
    (void)n_in; (void)out_size; (void)ws_size;
    const float* inputs = (const float*)d_in[0];
    const float* Wq     = (const float*)d_in[1];
    const float* bq     = (const float*)d_in[2];
    const float* Wk     = (const float*)d_in[3];
    const float* Wv     = (const float*)d_in[4];
    const float* Wo     = (const float*)d_in[5];
    const float* bo     = (const float*)d_in[6];
    const int*   src    = (const int*)d_in[7];
    const int*   dst    = (const int*)d_in[8];

    const int N = in_sizes[0] / HID;   // 50000
    const int E = in_sizes[7];         // 800000

    float* q  = (float*)d_ws;
    float* k  = q  + (size_t)N * HID;
    float* v  = k  + (size_t)N * HID;
    float* wv = v  + (size_t)N * HID;
    float* z  = wv + (size_t)N * HID;

    const int gemm_blocks = (N + 15) / 16;

    // fused Q/K/V projections (inputs staged in LDS once)
    qkv_proj_wmma<<<gemm_blocks, 256, 0, stream>>>(inputs, Wq, Wk, Wv, bq,
                                                   q, k, v, N);

    // zero accumulators (capture-safe memset nodes)
    hipMemsetAsync(wv, 0, (size_t)N * HID * sizeof(float), stream);
    hipMemsetAsync(z,  0, (size_t)N * NHEAD * sizeof(float), stream);

    // edge-wise attention scatter: E*8 wave32's, 8 per 256-thread block
    const long long npairs = (long long)E * NHEAD;
    const int edge_blocks = (int)((npairs * 32 + 255) / 256);
    edge_attn<<<edge_blocks, 256, 0, stream>>>(q, k, v, src, dst, wv, z, E);

    // normalize (fused) + output projection
    out_proj_wmma<<<gemm_blocks, 256, 0, stream>>>(wv, z, Wo, bo,
                                                   (float*)d_out, N);
}